// CMVN_35794257445186
// MI455X (gfx1250) — compile-verified
//
#include <hip/hip_runtime.h>
#include <stdint.h>

#define T_LEN   4096
#define F_DIM   120
#define B_DIM   64
#define BLOCK   256
#define NCHUNK  4          // float4 chunks per thread: 256 * 4 * 4 floats = 4096
#define EPSV    1e-10f

__global__ __launch_bounds__(BLOCK)
void cmvn_row_kernel(const float* __restrict__ x,
                     const int*   __restrict__ length,
                     float*       __restrict__ out)
{
    __shared__ float srow[T_LEN];   // staged row (16 KB)
    __shared__ float sred[16];      // [0..7] wave partials, [8] mean, [9] inv

    const int tid = threadIdx.x;
    const int row = blockIdx.x;                 // row = b * F + f  (C == 1)
    const int b   = row / F_DIM;
    const float* rowp = x   + (size_t)row * T_LEN;
    float*       outp = out + (size_t)row * T_LEN;

    // ---------------- async global -> LDS stage (CDNA5 DMA path) ----------
    // Each lane queues the same 4 x B128 chunks it will later read back, so a
    // per-wave s_wait_asynccnt 0 is sufficient (no workgroup barrier needed).
    const uint32_t lds_base = (uint32_t)(uintptr_t)(&srow[0]);
#pragma unroll
    for (int c = 0; c < NCHUNK; ++c) {
        uint32_t byte_off = (uint32_t)((c * BLOCK + tid) * 16);
        uint32_t lds_addr = lds_base + byte_off;
        asm volatile("global_load_async_to_lds_b128 %0, %1, %2 offset:0"
                     :: "v"(lds_addr), "v"(byte_off), "s"(rowp)
                     : "memory");
    }
    asm volatile("s_wait_asynccnt 0" ::: "memory");

    const int   len = length[b];
    const float n   = (float)len;

    // ---------------- pass 1: masked sum -> mean ---------------------------
    float4 v[NCHUNK];
    float  s = 0.0f;
#pragma unroll
    for (int c = 0; c < NCHUNK; ++c) {
        const int chunk = c * BLOCK + tid;          // float4 index in row
        v[c] = *reinterpret_cast<const float4*>(&srow[chunk * 4]);
        const int e = chunk * 4;
        s += (e + 0 < len) ? v[c].x : 0.0f;
        s += (e + 1 < len) ? v[c].y : 0.0f;
        s += (e + 2 < len) ? v[c].z : 0.0f;
        s += (e + 3 < len) ? v[c].w : 0.0f;
    }
#pragma unroll
    for (int off = 16; off > 0; off >>= 1) s += __shfl_xor(s, off, 32);
    if ((tid & 31) == 0) sred[tid >> 5] = s;
    __syncthreads();
    if (tid == 0) {
        float tot = 0.0f;
#pragma unroll
        for (int w = 0; w < BLOCK / 32; ++w) tot += sred[w];
        sred[8] = tot / n;
    }
    __syncthreads();
    const float mean = sred[8];

    // ---------------- pass 2: masked sum of squares -> 1/(eps+std) ---------
    float ss = 0.0f;
#pragma unroll
    for (int c = 0; c < NCHUNK; ++c) {
        const int e = (c * BLOCK + tid) * 4;
        float dx = v[c].x - mean, dy = v[c].y - mean;
        float dz = v[c].z - mean, dw = v[c].w - mean;
        ss += (e + 0 < len) ? dx * dx : 0.0f;
        ss += (e + 1 < len) ? dy * dy : 0.0f;
        ss += (e + 2 < len) ? dz * dz : 0.0f;
        ss += (e + 3 < len) ? dw * dw : 0.0f;
    }
#pragma unroll
    for (int off = 16; off > 0; off >>= 1) ss += __shfl_xor(ss, off, 32);
    if ((tid & 31) == 0) sred[tid >> 5] = ss;
    __syncthreads();
    if (tid == 0) {
        float tot = 0.0f;
#pragma unroll
        for (int w = 0; w < BLOCK / 32; ++w) tot += sred[w];
        const float var = tot / (n - 1.0f);
        sred[9] = 1.0f / (EPSV + sqrtf(var));
    }
    __syncthreads();
    const float inv = sred[9];

    // ---------------- pass 3: normalize ALL t, streamed B128 stores --------
#pragma unroll
    for (int c = 0; c < NCHUNK; ++c) {
        const int chunk = c * BLOCK + tid;
        float4 r;
        r.x = (v[c].x - mean) * inv;
        r.y = (v[c].y - mean) * inv;
        r.z = (v[c].z - mean) * inv;
        r.w = (v[c].w - mean) * inv;
        *reinterpret_cast<float4*>(&outp[chunk * 4]) = r;
    }
}

// Tuple output: (out, length) concatenated flat; d_out is float-typed, so the
// length tail is written as float-converted values.
__global__ void cmvn_len_tail_kernel(const int* __restrict__ length,
                                     float* __restrict__ tail, int n)
{
    int i = threadIdx.x + blockIdx.x * blockDim.x;
    if (i < n) tail[i] = (float)length[i];
}

extern "C" void kernel_launch(void* const* d_in, const int* in_sizes, int n_in,
                              void* d_out, int out_size, void* d_ws, size_t ws_size,
                              hipStream_t stream)
{
    const float* x      = (const float*)d_in[0];
    const int*   length = (const int*)d_in[1];
    float*       out    = (float*)d_out;

    const int rows = B_DIM * F_DIM;   // C == 1
    cmvn_row_kernel<<<rows, BLOCK, 0, stream>>>(x, length, out);
    cmvn_len_tail_kernel<<<1, B_DIM, 0, stream>>>(
        length, out + (size_t)rows * T_LEN, B_DIM);
}